// IoU_73985106641137
// MI455X (gfx1250) — compile-verified
//
#include <hip/hip_runtime.h>
#include <math.h>

#define BLK 256

typedef __attribute__((ext_vector_type(2))) float v2f;
typedef __attribute__((ext_vector_type(8))) float v8f;

// Deterministic wave32 sum using V_WMMA_F32_16X16X4_F32 with B = ones.
// A layout (32-bit A 16x4): lanes 0-15 vgpr0=K0, vgpr1=K1; lanes 16-31 vgpr0=K2, vgpr1=K3.
// Putting v in vgpr0 and 0 in vgpr1 gives row sums rowsum[m] = v[m] + v[m+16].
// D column-sum per lane then yields half-sums; one shfl_xor(16) completes the 32-lane sum.
// EXEC must be all-ones in the executing wave (call under `tid < 32` with full first wave).
__device__ __forceinline__ float wave32_sum_wmma(float v) {
  v2f a; a[0] = v;    a[1] = 0.0f;
  v2f b; b[0] = 1.0f; b[1] = 1.0f;
  v8f c = {0.f, 0.f, 0.f, 0.f, 0.f, 0.f, 0.f, 0.f};
  c = __builtin_amdgcn_wmma_f32_16x16x4_f32(
      /*neg_a=*/false, a, /*neg_b=*/false, b,
      /*c_mod=*/(short)0, c, /*reuse_a=*/false, /*reuse_b=*/false);
  float s = ((c[0] + c[1]) + (c[2] + c[3])) + ((c[4] + c[5]) + (c[6] + c[7]));
  s += __shfl_xor(s, 16, 32);
  return s;  // full 32-lane sum in every lane
}

__device__ __forceinline__ float shoelace4(const float x[4], const float y[4]) {
  float s = (x[0] * y[1] - x[1] * y[0]) + (x[1] * y[2] - x[2] * y[1]) +
            (x[2] * y[3] - x[3] * y[2]) + (x[3] * y[0] - x[0] * y[3]);
  return 0.5f * fabsf(s);
}

// Sort 4 points by atan2 angle about centroid (ascending), matching jnp.argsort.
__device__ __forceinline__ void sort4_by_angle(float x[4], float y[4]) {
  float cx = 0.25f * (x[0] + x[1] + x[2] + x[3]);
  float cy = 0.25f * (y[0] + y[1] + y[2] + y[3]);
  float ang[4];
#pragma unroll
  for (int i = 0; i < 4; ++i) ang[i] = atan2f(y[i] - cy, x[i] - cx);
#define CSWAP(i, j)                                                  \
  if (ang[i] > ang[j]) {                                             \
    float t_;                                                        \
    t_ = ang[i]; ang[i] = ang[j]; ang[j] = t_;                       \
    t_ = x[i];   x[i]   = x[j];   x[j]   = t_;                       \
    t_ = y[i];   y[i]   = y[j];   y[j]   = t_;                       \
  }
  CSWAP(0, 1) CSWAP(2, 3) CSWAP(0, 2) CSWAP(1, 3) CSWAP(1, 2)
#undef CSWAP
}

__global__ __launch_bounds__(BLK) void iou_pass1(const float* __restrict__ pred,
                                                 const float* __restrict__ targ,
                                                 float* __restrict__ partial, int B) {
  // Ping-pong polygon buffers in LDS. Layout [buf][point][tid]: stride between
  // points is 256 dwords == 0 mod 64 banks -> bank = tid mod 64, conflict-free
  // even with divergent per-thread point indices.
  __shared__ float sx[2][8][BLK];
  __shared__ float sy[2][8][BLK];
  __shared__ float red[BLK];

  const int t = threadIdx.x;
  const int idx = blockIdx.x * BLK + t;

  float v = 0.0f;
  if (idx < B) {
    // Coalesced 2x float4 per input per thread.
    const float4* p4 = (const float4*)(pred + (size_t)idx * 8);
    const float4* q4 = (const float4*)(targ + (size_t)idx * 8);
    float4 pa = p4[0], pb = p4[1];
    float4 qa = q4[0], qb = q4[1];
    float ax[4] = {pa.x, pa.z, pb.x, pb.z};
    float ay[4] = {pa.y, pa.w, pb.y, pb.w};
    float bx[4] = {qa.x, qa.z, qb.x, qb.z};
    float by[4] = {qa.y, qa.w, qb.y, qb.w};

    sort4_by_angle(ax, ay);
    sort4_by_angle(bx, by);

    const float area_a = shoelace4(ax, ay);
    const float area_b = shoelace4(bx, by);

    // Init subject polygon (pad with a[0] like the reference; pads add 0 area).
#pragma unroll
    for (int i = 0; i < 4; ++i) { sx[0][i][t] = ax[i]; sy[0][i][t] = ay[i]; }
#pragma unroll
    for (int i = 4; i < 8; ++i) { sx[0][i][t] = ax[0]; sy[0][i][t] = ay[0]; }

    int n = 4, ping = 0;
#pragma unroll
    for (int e = 0; e < 4; ++e) {
      const float c1x = bx[e], c1y = by[e];
      const float c2x = bx[(e + 1) & 3], c2y = by[(e + 1) & 3];
      const float ex = c2x - c1x, ey = c2y - c1y;
      const int q = ping ^ 1;
      int w = 0;
      for (int i = 0; i < n; ++i) {
        const int j = (i + 1 >= n) ? 0 : i + 1;
        const float cxp = sx[ping][i][t], cyp = sy[ping][i][t];
        const float nxp = sx[ping][j][t], nyp = sy[ping][j][t];
        const float dc = ex * (cyp - c1y) - ey * (cxp - c1x);
        const float dn = ex * (nyp - c1y) - ey * (nxp - c1x);
        const bool ci = (dc >= 0.0f), ni = (dn >= 0.0f);
        if (ci != ni) {
          const float den = dc - dn;
          const float tt = dc / ((den == 0.0f) ? 1.0f : den);
          sx[q][w][t] = cxp + tt * (nxp - cxp);
          sy[q][w][t] = cyp + tt * (nyp - cyp);
          ++w;
        }
        if (ni) { sx[q][w][t] = nxp; sy[q][w][t] = nyp; ++w; }
      }
      n = w;
      ping = q;
    }

    float inter = 0.0f;
    if (n > 0) {
      float s = 0.0f;
      for (int i = 0; i < n; ++i) {
        const int j = (i + 1 >= n) ? 0 : i + 1;
        s += sx[ping][i][t] * sy[ping][j][t] - sx[ping][j][t] * sy[ping][i][t];
      }
      inter = 0.5f * fabsf(s);
    }

    const float uni = area_a + area_b - inter;
    v = (uni > 0.0f) ? (inter / uni) : 0.0f;
  }

  // Deterministic fixed-order block reduction: LDS tree 256 -> 32, then WMMA.
  red[t] = v;
  __syncthreads();
  for (int s = BLK / 2; s >= 32; s >>= 1) {
    if (t < s) red[t] += red[t + s];
    __syncthreads();
  }
  if (t < 32) {
    const float tot = wave32_sum_wmma(red[t]);  // full first wave, EXEC all-ones
    if (t == 0) partial[blockIdx.x] = tot;
  }
}

__global__ __launch_bounds__(BLK) void iou_pass2(const float* __restrict__ partial,
                                                 int nblocks, float* __restrict__ out,
                                                 float invB) {
  __shared__ float red[BLK];
  const int t = threadIdx.x;
  float v = 0.0f;
  for (int i = t; i < nblocks; i += BLK) v += partial[i];  // fixed order per thread
  red[t] = v;
  __syncthreads();
  for (int s = BLK / 2; s >= 32; s >>= 1) {
    if (t < s) red[t] += red[t + s];
    __syncthreads();
  }
  if (t < 32) {
    const float tot = wave32_sum_wmma(red[t]);
    if (t == 0) out[0] = tot * invB;
  }
}

extern "C" void kernel_launch(void* const* d_in, const int* in_sizes, int n_in,
                              void* d_out, int out_size, void* d_ws, size_t ws_size,
                              hipStream_t stream) {
  const float* pred = (const float*)d_in[0];
  const float* targ = (const float*)d_in[1];
  const int B = in_sizes[0] / 8;  // (B, 4, 2) float32
  const int nblocks = (B + BLK - 1) / BLK;
  float* partial = (float*)d_ws;  // nblocks floats (~16 KB), fully overwritten each call

  iou_pass1<<<nblocks, BLK, 0, stream>>>(pred, targ, partial, B);
  iou_pass2<<<1, BLK, 0, stream>>>(partial, nblocks, (float*)d_out, 1.0f / (float)B);
}